// ResidualGCNLayer_36197984370746
// MI455X (gfx1250) — compile-verified
//
#include <hip/hip_runtime.h>

#define D 128
#define BN_EPS 1e-5f
#define LDS_STRIDE 260   // 256 + 4 pad floats -> lanes 0..15 hit distinct LDS banks

typedef __attribute__((ext_vector_type(2))) float v2f;
typedef __attribute__((ext_vector_type(8))) float v8f;

// ---------------------------------------------------------------------------
// Zero the workspace regions we accumulate into (ws is poisoned with 0xAA).
// ---------------------------------------------------------------------------
__global__ void rgcn_zero_kernel(float* __restrict__ p, size_t n) {
    size_t i = (size_t)blockIdx.x * blockDim.x + threadIdx.x;
    size_t stride = (size_t)gridDim.x * blockDim.x;
    for (; i < n; i += stride) p[i] = 0.0f;
}

// ---------------------------------------------------------------------------
// Edge scatter: one wave (32 lanes) per edge. Lane l moves float4 chunk l of
// x[src] into agg[dst] via 4 f32 atomics (consecutive lanes -> consecutive
// addresses of the same destination row => coalesced L2 atomics; agg fits in
// the 192MB L2). Lane 0 bumps the degree counter.
// ---------------------------------------------------------------------------
__global__ __launch_bounds__(256) void rgcn_scatter_kernel(
    const float* __restrict__ x,
    const long long* __restrict__ src,
    const long long* __restrict__ dst,
    float* __restrict__ agg,
    float* __restrict__ cnt,
    int E)
{
    const int lane = threadIdx.x & 31;
    const int e = blockIdx.x * (blockDim.x >> 5) + (threadIdx.x >> 5);
    if (e >= E) return;
    const long long s = src[e];
    const long long d = dst[e];
    const float4 v = ((const float4*)(x + (size_t)s * D))[lane];
    float* a = agg + (size_t)d * D + lane * 4;
    atomicAdd(a + 0, v.x);
    atomicAdd(a + 1, v.y);
    atomicAdd(a + 2, v.z);
    atomicAdd(a + 3, v.w);
    if (lane == 0) atomicAdd(cnt + d, 1.0f);
}

// ---------------------------------------------------------------------------
// Fused GEMM: pre = [mean_nb | x] (16x256) @ [W_l^T ; W_r^T] (256x128) + b_l
// using V_WMMA_F32_16X16X4_F32 (full f32, K=4 steps).
//
// Block = 16 node rows, 128 threads = 4 waves. Wave w computes N-tiles
// {32w..32w+15} and {32w+16..32w+31} with a shared A fragment per K-step.
//
// A fragment layout (32-bit A 16x4, ISA 7.12.2): lane L<16 -> row L, K={k,k+1};
// lane L>=16 -> row L-16, K={k+2,k+3}. A lives in LDS (stride 260 -> no bank
// conflicts). B fragment: B[k][n] = W[n][k] -> lane loads float2 W[n0+L%16][keff]
// straight from global (weights stay hot in L2/WGP$ across all 6250 blocks).
// ---------------------------------------------------------------------------
__global__ __launch_bounds__(128) void rgcn_gemm_kernel(
    const float* __restrict__ x,
    const float* __restrict__ agg,
    const float* __restrict__ cnt,
    const float* __restrict__ Wl,   // [D][D] row-major, Wl[n*D + k]
    const float* __restrict__ Wr,
    const float* __restrict__ bl,
    float* __restrict__ pre,        // [N][D] pre-BN output
    float* __restrict__ chsum,      // [D] per-channel sum
    float* __restrict__ chsq,       // [D] per-channel sum of squares
    int N)
{
    __shared__ float Acat[16 * LDS_STRIDE];

    const int tid  = threadIdx.x;
    const int wave = tid >> 5;
    const int lane = tid & 31;
    const int row0 = blockIdx.x * 16;

    // ---- Stage A: Acat[r][0..127] = agg/max(cnt,1), Acat[r][128..255] = x ----
    // 16 rows x 32 float4 per half = 512 float4 jobs per half; 4 per thread.
    #pragma unroll
    for (int it = 0; it < 4; ++it) {
        int idx  = it * 128 + tid;      // 0..511
        int r    = idx >> 5;            // row 0..15
        int c4   = idx & 31;            // float4 column 0..31
        int node = row0 + r;
        if (node >= N) node = N - 1;    // clamp (N % 16 == 0 in practice)
        const float4 xv = ((const float4*)(x   + (size_t)node * D))[c4];
        const float4 av = ((const float4*)(agg + (size_t)node * D))[c4];
        const float inv = 1.0f / fmaxf(cnt[node], 1.0f);
        float* arow = Acat + r * LDS_STRIDE;
        arow[c4 * 4 + 0] = av.x * inv;
        arow[c4 * 4 + 1] = av.y * inv;
        arow[c4 * 4 + 2] = av.z * inv;
        arow[c4 * 4 + 3] = av.w * inv;
        arow[128 + c4 * 4 + 0] = xv.x;
        arow[128 + c4 * 4 + 1] = xv.y;
        arow[128 + c4 * 4 + 2] = xv.z;
        arow[128 + c4 * 4 + 3] = xv.w;
    }
    __syncthreads();

    // ---- WMMA K-loop ----
    v8f acc0 = {};   // N-tile at n0
    v8f acc1 = {};   // N-tile at n0+16
    const int n0   = wave * 32;
    const int kh   = (lane >> 4) * 2;   // lane K sub-offset: 0 or 2
    const int nlan = lane & 15;         // B column within tile / A row

    const float* Ap  = Acat + nlan * LDS_STRIDE + kh;
    const float* B0l = Wl + (size_t)(n0 + nlan)      * D + kh;
    const float* B1l = Wl + (size_t)(n0 + 16 + nlan) * D + kh;
    const float* B0r = Wr + (size_t)(n0 + nlan)      * D + kh;
    const float* B1r = Wr + (size_t)(n0 + 16 + nlan) * D + kh;

    #pragma unroll 8
    for (int k = 0; k < 128; k += 4) {          // mean_nb @ W_l^T
        v2f a  = *(const v2f*)(Ap + k);
        v2f b0 = *(const v2f*)(B0l + k);
        v2f b1 = *(const v2f*)(B1l + k);
        acc0 = __builtin_amdgcn_wmma_f32_16x16x4_f32(false, a, false, b0,
                                                     (short)0, acc0, false, false);
        acc1 = __builtin_amdgcn_wmma_f32_16x16x4_f32(false, a, false, b1,
                                                     (short)0, acc1, false, false);
    }
    #pragma unroll 8
    for (int k = 0; k < 128; k += 4) {          // x @ W_r^T
        v2f a  = *(const v2f*)(Ap + 128 + k);
        v2f b0 = *(const v2f*)(B0r + k);
        v2f b1 = *(const v2f*)(B1r + k);
        acc0 = __builtin_amdgcn_wmma_f32_16x16x4_f32(false, a, false, b0,
                                                     (short)0, acc0, false, false);
        acc1 = __builtin_amdgcn_wmma_f32_16x16x4_f32(false, a, false, b1,
                                                     (short)0, acc1, false, false);
    }

    // ---- Epilogue: bias, store pre-BN, per-channel stats ----
    // C/D layout (ISA 7.12.2): VGPR r -> M = r (lanes 0-15) / r+8 (lanes 16-31),
    // N = lane % 16.
    const int rbase = (lane >> 4) * 8;
    {
        const int col = n0 + nlan;
        const float bias = bl[col];
        float s = 0.0f, q = 0.0f;
        #pragma unroll
        for (int r = 0; r < 8; ++r) {
            const int node = row0 + rbase + r;
            const float v = acc0[r] + bias;
            if (node < N) pre[(size_t)node * D + col] = v;
            s += v; q += v * v;
        }
        atomicAdd(chsum + col, s);
        atomicAdd(chsq  + col, q);
    }
    {
        const int col = n0 + 16 + nlan;
        const float bias = bl[col];
        float s = 0.0f, q = 0.0f;
        #pragma unroll
        for (int r = 0; r < 8; ++r) {
            const int node = row0 + rbase + r;
            const float v = acc1[r] + bias;
            if (node < N) pre[(size_t)node * D + col] = v;
            s += v; q += v * v;
        }
        atomicAdd(chsum + col, s);
        atomicAdd(chsq  + col, q);
    }
}

// ---------------------------------------------------------------------------
// BatchNorm (batch stats) + ReLU + identity residual.
// ---------------------------------------------------------------------------
__global__ __launch_bounds__(256) void rgcn_finalize_kernel(
    const float* __restrict__ pre,
    const float* __restrict__ x,
    const float* __restrict__ chsum,
    const float* __restrict__ chsq,
    const float* __restrict__ gamma,
    const float* __restrict__ beta,
    float* __restrict__ out,
    int N)
{
    const size_t total = (size_t)N * D;
    const float invN = 1.0f / (float)N;
    size_t i = (size_t)blockIdx.x * blockDim.x + threadIdx.x;
    size_t stride = (size_t)gridDim.x * blockDim.x;
    for (; i < total; i += stride) {
        const int col = (int)(i & (D - 1));
        const float mu  = chsum[col] * invN;
        const float var = chsq[col] * invN - mu * mu;
        float y = gamma[col] * (pre[i] - mu) * rsqrtf(var + BN_EPS) + beta[col];
        y = fmaxf(y, 0.0f);
        out[i] = y + x[i];
    }
}

// ---------------------------------------------------------------------------
extern "C" void kernel_launch(void* const* d_in, const int* in_sizes, int n_in,
                              void* d_out, int out_size, void* d_ws, size_t ws_size,
                              hipStream_t stream) {
    const float*     x     = (const float*)d_in[0];
    const long long* ei    = (const long long*)d_in[1];   // int64 edge_index [2][E]
    const float*     Wl    = (const float*)d_in[2];
    const float*     bl    = (const float*)d_in[3];
    const float*     Wr    = (const float*)d_in[4];
    const float*     gamma = (const float*)d_in[5];
    const float*     beta  = (const float*)d_in[6];
    float*           out   = (float*)d_out;

    const int N = in_sizes[0] / D;
    const int E = in_sizes[1] / 2;
    const long long* src = ei;
    const long long* dst = ei + E;

    // Workspace layout
    float* agg   = (float*)d_ws;                 // N*D
    float* cnt   = agg + (size_t)N * D;          // N
    float* pre   = cnt + N;                      // N*D
    float* chsum = pre + (size_t)N * D;          // D
    float* chsq  = chsum + D;                    // D

    // 1) zero accumulators (agg, cnt, chsum, chsq) -- 'pre' is fully overwritten
    {
        size_t nz1 = (size_t)N * D + N;          // agg + cnt (contiguous)
        rgcn_zero_kernel<<<2048, 256, 0, stream>>>(agg, nz1);
        rgcn_zero_kernel<<<1, 256, 0, stream>>>(chsum, (size_t)(2 * D));
    }

    // 2) edge scatter-add (one wave per edge)
    {
        const int waves_per_block = 256 / 32;
        const int blocks = (E + waves_per_block - 1) / waves_per_block;
        rgcn_scatter_kernel<<<blocks, 256, 0, stream>>>(x, src, dst, agg, cnt, E);
    }

    // 3) fused dual-GEMM via f32 WMMA + channel stats
    {
        const int blocks = (N + 15) / 16;
        rgcn_gemm_kernel<<<blocks, 128, 0, stream>>>(x, agg, cnt, Wl, Wr, bl,
                                                     pre, chsum, chsq, N);
    }

    // 4) BN + ReLU + residual
    rgcn_finalize_kernel<<<2048, 256, 0, stream>>>(pre, x, chsum, chsq,
                                                   gamma, beta, out, N);
}